// TJDLayer_59227599011917
// MI455X (gfx1250) — compile-verified
//
#include <hip/hip_runtime.h>
#include <hip/hip_bf16.h>
#include <math.h>

// Problem constants (from reference)
#define EMB   1024
#define RANK  4
#define VOCAB 512
#define BB    8
#define TT    1024
#define NCOL  (VOCAB * RANK * RANK)   // 8192 columns of w_vocab

typedef float v2f __attribute__((ext_vector_type(2)));
typedef float v8f __attribute__((ext_vector_type(8)));

// ---------------------------------------------------------------------------
// Kernel 1: xbar[row][d] = mean over T of x[row,:,d]  (rows 8..15 zero-padded
// so the GEMM M-dimension is a full 16). Pure bandwidth: 33.5 MB in,
// float4-vectorized -> global_load_b128, 4x fewer VMEM issues.
// grid = (1, 16), block = 256: each thread owns one float4 column of a row.
// ---------------------------------------------------------------------------
__global__ void __launch_bounds__(256) k_mean(const float* __restrict__ x,
                                              float* __restrict__ xbar) {
    const int tid = threadIdx.x;          // float4 column 0..255
    const int row = blockIdx.y;
    float4 s = make_float4(0.f, 0.f, 0.f, 0.f);
    if (row < BB) {
        const float* p = x + (size_t)row * TT * EMB + tid * 4;
#pragma unroll 4
        for (int t = 0; t < TT; ++t) {
            const float4 v = *(const float4*)(p + (size_t)t * EMB);
            s.x += v.x; s.y += v.y; s.z += v.z; s.w += v.w;
        }
        const float inv = 1.0f / TT;
        s.x *= inv; s.y *= inv; s.z *= inv; s.w *= inv;
    }
    ((float4*)xbar)[row * (EMB / 4) + tid] = s;
}

// ---------------------------------------------------------------------------
// Kernel 2: core[8][8192] = xbar16[16][1024] @ w_vocab[1024][8192]
// using V_WMMA_F32_16X16X4_F32. One 16x16 output tile per wave, K=1024 loop,
// software-pipelined: fragments for step i+1 are in flight while the WMMA for
// step i executes. A panel staged in LDS as [k][m] (conflict-free, pairs
// fold into ds_load_2addr_b32). Unconditional speculative prefetch of the
// weight stream (translation failures are silently dropped per ISA).
// grid = 64 blocks x 256 threads (8 waves) -> 512 tiles = N/16.
// ---------------------------------------------------------------------------
__global__ void __launch_bounds__(256) k_gemm(const float* __restrict__ xbar,
                                              const float* __restrict__ Wv,
                                              float* __restrict__ core) {
    __shared__ float sA[16 * EMB];            // 64 KB, [k][m] layout
    // Coalesced global read of the A panel, transposed write into LDS.
    for (int idx = threadIdx.x; idx < 16 * EMB; idx += 256) {
        const int m = idx >> 10;              // row (padded batch)
        const int k = idx & (EMB - 1);        // K index
        sA[k * 16 + m] = xbar[idx];
    }
    __syncthreads();

    const int lane = threadIdx.x & 31;
    const int wave = threadIdx.x >> 5;
    const int n0   = (blockIdx.x * 8 + wave) * 16;   // tile column base
    const int m    = lane & 15;
    const int koff = (lane >> 4) * 2;                // ISA A/B layout: K 0,1 | 2,3
    const int n    = n0 + m;

    v8f acc = {};

    // Prologue: load fragments for kk = 0.
    v2f a, b;
    a.x = sA[koff * 16 + m];
    a.y = sA[(koff + 1) * 16 + m];
    b.x = Wv[(size_t)koff * NCOL + n];
    b.y = Wv[(size_t)(koff + 1) * NCOL + n];

#pragma unroll 4
    for (int kk = 4; kk < EMB; kk += 4) {
        // Speculative prefetch ~3 steps ahead of the weight stream.
        __builtin_prefetch(&Wv[(size_t)(kk + 12 + koff) * NCOL + n], 0, 2);

        // Issue next-step fragment loads before consuming the current ones.
        v2f an, bn;
        an.x = sA[(kk + koff)     * 16 + m];
        an.y = sA[(kk + koff + 1) * 16 + m];
        bn.x = Wv[(size_t)(kk + koff)     * NCOL + n];
        bn.y = Wv[(size_t)(kk + koff + 1) * NCOL + n];

        // 8 args: (neg_a, A, neg_b, B, c_mod, C, reuse_a, reuse_b)
        acc = __builtin_amdgcn_wmma_f32_16x16x4_f32(
            false, a, false, b, (short)0, acc, false, false);

        a = an; b = bn;
    }
    // Epilogue: last step.
    acc = __builtin_amdgcn_wmma_f32_16x16x4_f32(
        false, a, false, b, (short)0, acc, false, false);

    // C/D layout: VGPR i holds M=i (lanes 0-15) and M=i+8 (lanes 16-31).
    // Rows 8..15 are zero padding -> only lanes 0-15 store.
    if (lane < 16) {
#pragma unroll
        for (int i = 0; i < 8; ++i)
            core[(size_t)i * NCOL + n0 + lane] = acc[i];
    }
}

// 4x4 matmul helper: D = A * B (row-major, flat 16 floats)
__device__ __forceinline__ void mm4(const float* A, const float* B, float* D) {
#pragma unroll
    for (int i = 0; i < 4; ++i)
#pragma unroll
        for (int j = 0; j < 4; ++j) {
            float s = 0.0f;
#pragma unroll
            for (int k = 0; k < 4; ++k) s += A[i * 4 + k] * B[k * 4 + j];
            D[i * 4 + j] = s;
        }
}

// ---------------------------------------------------------------------------
// Kernel 3: per batch b (grid=8, block=256):
//  - alpha/beta = xbar[b] @ w_alpha/w_beta, normalized to prob. vectors
//  - gather label cores, column-normalize each 4x4
//  - associative tree-product of the 1024-matrix chain (8 LDS levels)
//  - lossb[b] = -log(beta_n . (prod C_t) . alpha_n)
// ---------------------------------------------------------------------------
__global__ void __launch_bounds__(256) k_chain(const float* __restrict__ xbar,
                                               const float* __restrict__ core,
                                               const int*   __restrict__ labels,
                                               const float* __restrict__ wA,
                                               const float* __restrict__ wB,
                                               float* __restrict__ lossb) {
    const int b   = blockIdx.x;
    const int tid = threadIdx.x;
    __shared__ float red[256][8];
    __shared__ float Msh[256][16];
    __shared__ float sna[4], snb[4];

    // ---- alpha / beta projections (xbar already includes the 1/T mean) ----
    float pa[4] = {0, 0, 0, 0}, pb[4] = {0, 0, 0, 0};
    for (int d = tid; d < EMB; d += 256) {
        const float xv = xbar[b * EMB + d];
#pragma unroll
        for (int r = 0; r < 4; ++r) {
            pa[r] += xv * wA[d * 4 + r];
            pb[r] += xv * wB[d * 4 + r];
        }
    }
#pragma unroll
    for (int r = 0; r < 4; ++r) { red[tid][r] = pa[r]; red[tid][4 + r] = pb[r]; }
    __syncthreads();
    for (int s = 128; s > 0; s >>= 1) {
        if (tid < s) {
#pragma unroll
            for (int r = 0; r < 8; ++r) red[tid][r] += red[tid + s][r];
        }
        __syncthreads();
    }
    if (tid == 0) {
        float sa = 0.0f, sb = 0.0f;
#pragma unroll
        for (int r = 0; r < 4; ++r) { sa += fabsf(red[0][r]); sb += fabsf(red[0][4 + r]); }
#pragma unroll
        for (int r = 0; r < 4; ++r) {
            sna[r] = fabsf(red[0][r]) / sa;
            snb[r] = fabsf(red[0][4 + r]) / sb;
        }
    }

    // ---- local product of 4 consecutive normalized core matrices ----------
    float M[16];
#pragma unroll
    for (int step = 0; step < 4; ++step) {
        const int t   = tid * 4 + step;
        const int lbl = labels[b * TT + t];
        const float* cp = core + (size_t)b * NCOL + lbl * 4;  // (i,j) at i*2048 + j
        float c[4][4];
#pragma unroll
        for (int i = 0; i < 4; ++i) {
            const float4 rv = *(const float4*)(cp + i * 2048);
            c[i][0] = fabsf(rv.x); c[i][1] = fabsf(rv.y);
            c[i][2] = fabsf(rv.z); c[i][3] = fabsf(rv.w);
        }
        float nc[16];
#pragma unroll
        for (int j = 0; j < 4; ++j) {
            const float cs = c[0][j] + c[1][j] + c[2][j] + c[3][j];
            const float inv = 1.0f / cs;
#pragma unroll
            for (int i = 0; i < 4; ++i) nc[i * 4 + j] = c[i][j] * inv;
        }
        if (step == 0) {
#pragma unroll
            for (int i = 0; i < 16; ++i) M[i] = nc[i];
        } else {
            float Nn[16];
            mm4(nc, M, Nn);  // later step multiplies on the left
#pragma unroll
            for (int i = 0; i < 16; ++i) M[i] = Nn[i];
        }
    }
#pragma unroll
    for (int i = 0; i < 16; ++i) Msh[tid][i] = M[i];

    // ---- tree reduction of the chain: combine = later @ earlier -----------
    for (int off = 1; off < 256; off <<= 1) {
        __syncthreads();
        if ((tid & (2 * off - 1)) == 0) {
            float L[16], Nn[16];
#pragma unroll
            for (int i = 0; i < 16; ++i) L[i] = Msh[tid + off][i];
            mm4(L, M, Nn);
#pragma unroll
            for (int i = 0; i < 16; ++i) { M[i] = Nn[i]; Msh[tid][i] = Nn[i]; }
        }
    }
    __syncthreads();

    if (tid == 0) {
        float v[4];
#pragma unroll
        for (int i = 0; i < 4; ++i) {
            float s = 0.0f;
#pragma unroll
            for (int j = 0; j < 4; ++j) s += M[i * 4 + j] * sna[j];
            v[i] = s;
        }
        float p = 0.0f;
#pragma unroll
        for (int i = 0; i < 4; ++i) p += snb[i] * v[i];
        lossb[b] = -logf(p);
    }
}

// ---------------------------------------------------------------------------
// Kernel 4: mean over batches -> scalar output
// ---------------------------------------------------------------------------
__global__ void k_final(const float* __restrict__ lossb, float* __restrict__ out) {
    float s = 0.0f;
#pragma unroll
    for (int i = 0; i < BB; ++i) s += lossb[i];
    out[0] = s * (1.0f / BB);
}

// ---------------------------------------------------------------------------
extern "C" void kernel_launch(void* const* d_in, const int* in_sizes, int n_in,
                              void* d_out, int out_size, void* d_ws, size_t ws_size,
                              hipStream_t stream) {
    const float* x      = (const float*)d_in[0];   // [8,1024,1024] f32
    const int*   labels = (const int*)d_in[1];     // [8,1024] int
    const float* wA     = (const float*)d_in[2];   // [1024,4] f32
    const float* wB     = (const float*)d_in[3];   // [1024,4] f32
    const float* Wv     = (const float*)d_in[4];   // [1024,8192] f32
    float* out = (float*)d_out;

    float* xbar  = (float*)d_ws;                   // 16*1024 floats (64 KB)
    float* core  = xbar + 16 * EMB;                // 8*8192 floats (256 KB)
    float* lossb = core + BB * NCOL;               // 8 floats

    k_mean <<<dim3(1, 16), 256, 0, stream>>>(x, xbar);
    k_gemm <<<dim3(64),    256, 0, stream>>>(xbar, Wv, core);
    k_chain<<<dim3(BB),    256, 0, stream>>>(xbar, core, labels, wA, wB, lossb);
    k_final<<<1, 1, 0, stream>>>(lossb, out);
}